// RelationGatedGNN_76055280877921
// MI455X (gfx1250) — compile-verified
//
#include <hip/hip_runtime.h>
#include <hip/hip_bf16.h>

#define NN 50000
#define EE 800000
#define HD 64
#define RR 8

typedef float v2f __attribute__((ext_vector_type(2)));
typedef float v8f __attribute__((ext_vector_type(8)));

// ---------------- WMMA helpers (V_WMMA_F32_16X16X4_F32, fp32 path) ----------
// A (16x4 f32): lane = {m = lane&15, Kpair = 2*(lane>>4)}, .x=K, .y=K+1
// B (4x16 f32): lane = {n = lane&15, Kpair = 2*(lane>>4)}, .x=K, .y=K+1
// C/D (16x16 f32): acc[v] at (row = v + 8*(lane>>4), col = lane&15)

struct AFrag { v2f a[16]; };

__device__ __forceinline__ AFrag load_afrag(const float* __restrict__ A, int nrow0) {
    int lane = threadIdx.x & 31;
    int half = lane >> 4;
    int i16  = lane & 15;
    const float* arow = A + (size_t)(nrow0 + i16) * HD;
    AFrag f;
#pragma unroll
    for (int t = 0; t < 16; ++t)
        f.a[t] = *(const v2f*)(arow + 4 * t + 2 * half);
    return f;
}

__device__ __forceinline__ v8f gemm_afrag(const AFrag& af, const float* __restrict__ W,
                                          int jbase, v8f acc) {
    int lane = threadIdx.x & 31;
    int half = lane >> 4;
    int i16  = lane & 15;
    const float* brow = W + (size_t)(jbase + i16) * HD;
#pragma unroll
    for (int t = 0; t < 16; ++t) {
        v2f b = *(const v2f*)(brow + 4 * t + 2 * half);
        acc = __builtin_amdgcn_wmma_f32_16x16x4_f32(false, af.a[t], false, b,
                                                    (short)0, acc, false, false);
    }
    return acc;
}

// ---------------- float <-> orderable uint for atomic max ------------------
__device__ __forceinline__ unsigned enc_f(float f) {
    unsigned b = __float_as_uint(f);
    return (b & 0x80000000u) ? ~b : (b | 0x80000000u);
}
__device__ __forceinline__ float dec_f(unsigned u) {
    unsigned b = (u & 0x80000000u) ? (u & 0x7fffffffu) : ~u;
    return __uint_as_float(b);
}

// ---------------- dense 64x64 GEMM (+bias, optional relu) ------------------
template <bool RELU>
__global__ void k_dense64(const float* __restrict__ A, const float* __restrict__ W,
                          const float* __restrict__ bias, float* __restrict__ out) {
    int nrow0 = blockIdx.x * 16;
    int jbase = (threadIdx.x >> 5) * 16;
    AFrag af = load_afrag(A, nrow0);
    v8f acc = {0.f, 0.f, 0.f, 0.f, 0.f, 0.f, 0.f, 0.f};
    acc = gemm_afrag(af, W, jbase, acc);
    int lane = threadIdx.x & 31, half = lane >> 4, i16 = lane & 15;
    int col = jbase + i16;
    float bv = bias[col];
#pragma unroll
    for (int v = 0; v < 8; ++v) {
        int row = nrow0 + v + 8 * half;
        float val = acc[v] + bv;
        if (RELU) val = fmaxf(val, 0.f);
        out[(size_t)row * HD + col] = val;
    }
}

// ---------------- hWr[n][r][k] = sum_h h[n,h] * W_rel[r,k,h] ---------------
__global__ void k_hwr(const float* __restrict__ h, const float* __restrict__ W_rel,
                      float* __restrict__ hWr) {
    int nrow0 = blockIdx.x * 16;
    int jbase = (threadIdx.x >> 5) * 16;
    AFrag af = load_afrag(h, nrow0);
    int lane = threadIdx.x & 31, half = lane >> 4, i16 = lane & 15;
    int col = jbase + i16;
    for (int r = 0; r < RR; ++r) {
        v8f acc = {0.f, 0.f, 0.f, 0.f, 0.f, 0.f, 0.f, 0.f};
        acc = gemm_afrag(af, W_rel + (size_t)r * HD * HD, jbase, acc);
#pragma unroll
        for (int v = 0; v < 8; ++v) {
            int row = nrow0 + v + 8 * half;
            hWr[((size_t)row * RR + r) * HD + col] = acc[v];
        }
    }
}

// -------- gate GEMM fused: h_new = mean_r sigmoid(h@Wg[r].T + bg[r]) * agg --
__global__ void k_gate(const float* __restrict__ h, const float* __restrict__ W_gate,
                       const float* __restrict__ b_gate, const float* __restrict__ agg,
                       float* __restrict__ h_new) {
    int nrow0 = blockIdx.x * 16;
    int jbase = (threadIdx.x >> 5) * 16;
    AFrag af = load_afrag(h, nrow0);
    int lane = threadIdx.x & 31, half = lane >> 4, i16 = lane & 15;
    int col = jbase + i16;
    float hn[8] = {0.f, 0.f, 0.f, 0.f, 0.f, 0.f, 0.f, 0.f};
    for (int r = 0; r < RR; ++r) {
        v8f acc = {0.f, 0.f, 0.f, 0.f, 0.f, 0.f, 0.f, 0.f};
        acc = gemm_afrag(af, W_gate + (size_t)r * HD * HD, jbase, acc);
        float bg = b_gate[r * HD + col];
#pragma unroll
        for (int v = 0; v < 8; ++v) {
            int row = nrow0 + v + 8 * half;
            float g = 1.f / (1.f + __expf(-(acc[v] + bg)));
            hn[v] += g * agg[((size_t)row * RR + r) * HD + col];
        }
    }
#pragma unroll
    for (int v = 0; v < 8; ++v) {
        int row = nrow0 + v + 8 * half;
        h_new[(size_t)row * HD + col] = hn[v] * (1.f / RR);
    }
}

// ---------------- per-node attention projections S,T (N x R) ---------------
__global__ void k_st(const float* __restrict__ h, const float* __restrict__ W_att,
                     float* __restrict__ S, float* __restrict__ T) {
    __shared__ float watt[RR * 2 * HD];  // 4 KB
    for (int i = threadIdx.x; i < RR * 2 * HD; i += 256) watt[i] = W_att[i];
    __syncthreads();
    int n = blockIdx.x * 256 + threadIdx.x;
    if (n >= NN) return;
    float Ss[RR] = {0.f}, Ts[RR] = {0.f};
    const float* hr = h + (size_t)n * HD;
    for (int k = 0; k < HD; ++k) {
        float hv = hr[k];
#pragma unroll
        for (int r = 0; r < RR; ++r) {
            Ss[r] += hv * watt[r * 2 * HD + k];
            Ts[r] += hv * watt[r * 2 * HD + HD + k];
        }
    }
#pragma unroll
    for (int r = 0; r < RR; ++r) {
        S[(size_t)n * RR + r] = Ss[r];
        T[(size_t)n * RR + r] = Ts[r];
    }
}

// ------------- edge scores + leaky relu + global per-relation max ----------
__global__ void k_score(const float* __restrict__ S, const float* __restrict__ T,
                        const int* __restrict__ edge_index, const int* __restrict__ edge_type,
                        const float* __restrict__ b_att, float* __restrict__ scores,
                        unsigned* __restrict__ m_enc) {
    __shared__ unsigned smax[RR];
    int tid = threadIdx.x;
    if (tid < RR) smax[tid] = 0u;
    __syncthreads();
    int e = blockIdx.x * 256 + tid;
    int et  = edge_type[e];
    int src = edge_index[e];
    int dst = edge_index[EE + e];
    float s = S[(size_t)src * RR + et] + T[(size_t)dst * RR + et] + b_att[et];
    s = (s >= 0.f) ? s : 0.2f * s;  // leaky relu
    scores[e] = s;
    atomicMax(&smax[et], enc_f(s));
    __syncthreads();
    if (tid < RR) atomicMax(&m_enc[tid], smax[tid]);
}

// ------------- ex = exp(score - m[et]); denom[r] = sum ex ------------------
__global__ void k_ex(const float* __restrict__ scores, const int* __restrict__ edge_type,
                     const unsigned* __restrict__ m_enc, float* __restrict__ denom,
                     float* __restrict__ exb) {
    __shared__ float sden[RR];
    int tid = threadIdx.x;
    if (tid < RR) sden[tid] = 0.f;
    __syncthreads();
    int e = blockIdx.x * 256 + tid;
    int et = edge_type[e];
    float ex = __expf(scores[e] - dec_f(m_enc[et]));
    exb[e] = ex;
    atomicAdd(&sden[et], ex);
    __syncthreads();
    if (tid < RR) atomicAdd(&denom[tid], sden[tid]);
}

// ------------- scatter: agg[dst,et,:] += hWr[src,et,:] * att ---------------
__global__ void k_msg(const float* __restrict__ hWr, const float* __restrict__ exb,
                      const float* __restrict__ denom, const int* __restrict__ edge_index,
                      const int* __restrict__ edge_type, float* __restrict__ agg) {
    int t = blockIdx.x * 256 + threadIdx.x;  // E*64 threads
    int e = t >> 6;
    int c = t & 63;
    int et  = edge_type[e];
    float att = exb[e] / denom[et];
    int src = edge_index[e];
    int dst = edge_index[EE + e];
    float val = hWr[((size_t)src * RR + et) * HD + c] * att;
    atomicAdd(&agg[((size_t)dst * RR + et) * HD + c], val);
}

// --------- h = relu(LN(h + h_new)); one wave32 per node, 2 elems/lane ------
__global__ void k_ln(float* __restrict__ h, const float* __restrict__ h_new,
                     const float* __restrict__ gamma, const float* __restrict__ beta) {
    int n = blockIdx.x * 8 + (threadIdx.x >> 5);
    int lane = threadIdx.x & 31;
    size_t base = (size_t)n * HD;
    float a0 = h[base + lane]      + h_new[base + lane];
    float a1 = h[base + lane + 32] + h_new[base + lane + 32];
    float s = a0 + a1;
#pragma unroll
    for (int off = 16; off; off >>= 1) s += __shfl_xor(s, off, 32);
    float mean = s * (1.f / HD);
    float d0 = a0 - mean, d1 = a1 - mean;
    float q = d0 * d0 + d1 * d1;
#pragma unroll
    for (int off = 16; off; off >>= 1) q += __shfl_xor(q, off, 32);
    float rstd = rsqrtf(q * (1.f / HD) + 1e-5f);
    float y0 = fmaxf(d0 * rstd * gamma[lane]      + beta[lane],      0.f);
    float y1 = fmaxf(d1 * rstd * gamma[lane + 32] + beta[lane + 32], 0.f);
    h[base + lane]      = y0;
    h[base + lane + 32] = y1;
}

// ---------------------------------------------------------------------------
extern "C" void kernel_launch(void* const* d_in, const int* in_sizes, int n_in,
                              void* d_out, int out_size, void* d_ws, size_t ws_size,
                              hipStream_t stream) {
    const float* x        = (const float*)d_in[0];
    const int*   edge_idx = (const int*)  d_in[1];
    const int*   edge_ty  = (const int*)  d_in[2];
    const float* W_in     = (const float*)d_in[3];
    const float* b_in     = (const float*)d_in[4];
    const float* W_rel    = (const float*)d_in[5];
    const float* W_gate   = (const float*)d_in[6];
    const float* b_gate   = (const float*)d_in[7];
    const float* W_att    = (const float*)d_in[8];
    const float* b_att    = (const float*)d_in[9];
    const float* ln_gamma = (const float*)d_in[10];
    const float* ln_beta  = (const float*)d_in[11];
    const float* W_out    = (const float*)d_in[12];
    const float* b_out    = (const float*)d_in[13];
    float* out = (float*)d_out;

    float* ws = (float*)d_ws;
    size_t off = 0;
    float* h      = ws + off; off += (size_t)NN * HD;
    float* h_new  = ws + off; off += (size_t)NN * HD;
    float* hWr    = ws + off; off += (size_t)NN * RR * HD;
    float* agg    = ws + off; off += (size_t)NN * RR * HD;
    float* S      = ws + off; off += (size_t)NN * RR;
    float* T      = ws + off; off += (size_t)NN * RR;
    float* scores = ws + off; off += (size_t)EE;
    float* exb    = ws + off; off += (size_t)EE;
    float* denom  = ws + off; off += RR;
    unsigned* m_enc = (unsigned*)(ws + off); off += RR;

    const int gTiles = NN / 16;        // 3125, exact
    const int gEdge  = EE / 256;       // 3125, exact
    const int gMsg   = (EE * 64) / 256; // 200000, exact
    const int gLn    = NN / 8;         // 6250, exact

    // h = relu(x @ W_in.T + b_in)
    k_dense64<true><<<gTiles, 128, 0, stream>>>(x, W_in, b_in, h);

    for (int layer = 0; layer < 3; ++layer) {
        hipMemsetAsync(agg,   0, (size_t)NN * RR * HD * sizeof(float), stream);
        hipMemsetAsync(m_enc, 0, RR * sizeof(unsigned), stream);
        hipMemsetAsync(denom, 0, RR * sizeof(float), stream);

        k_st   <<<(NN + 255) / 256, 256, 0, stream>>>(h, W_att, S, T);
        k_score<<<gEdge, 256, 0, stream>>>(S, T, edge_idx, edge_ty, b_att, scores, m_enc);
        k_ex   <<<gEdge, 256, 0, stream>>>(scores, edge_ty, m_enc, denom, exb);
        k_hwr  <<<gTiles, 128, 0, stream>>>(h, W_rel, hWr);
        k_msg  <<<gMsg,  256, 0, stream>>>(hWr, exb, denom, edge_idx, edge_ty, agg);
        k_gate <<<gTiles, 128, 0, stream>>>(h, W_gate, b_gate, agg, h_new);
        k_ln   <<<gLn,   256, 0, stream>>>(h, h_new, ln_gamma + layer * HD, ln_beta + layer * HD);
    }

    // out = h @ W_out.T + b_out
    k_dense64<false><<<gTiles, 128, 0, stream>>>(h, W_out, b_out, out);
}